// IterTranslatedSpatialCorrelationSampler_62130996904371
// MI455X (gfx1250) — compile-verified
//
#include <hip/hip_runtime.h>

typedef __attribute__((ext_vector_type(2))) float v2f;
typedef __attribute__((ext_vector_type(8))) float v8f;

#define BB 8
#define CC 256
#define HH 64
#define WW 128
#define HW (HH * WW)
#define NPATCH 9   // 9x9 offsets
#define NWIN 10    // 10x10 integer window per pixel

// One wave (32 lanes) per output pixel; 8 waves per block.
// Per-wave LDS: 256 floats (in1 pixel vector) + 256 floats (D / T scratch).
__global__ __launch_bounds__(256) void corr_flow_kernel(
    const float* __restrict__ in1,
    const float* __restrict__ in2,
    const float* __restrict__ flow,
    float* __restrict__ out)
{
    __shared__ __align__(16) float smem[8 * 512];

    const int lane = threadIdx.x & 31;
    const int wave = threadIdx.x >> 5;
    const int pix  = blockIdx.x * 8 + wave;

    const int b  = pix / HW;
    const int hw = pix - b * HW;
    const int hi = hw / WW;
    const int wi = hw - hi * WW;

    // ---- base sampling position & shared bilinear weights -----------------
    const float fx = flow[((size_t)b * 2 + 0) * HW + hw];
    const float fy = flow[((size_t)b * 2 + 1) * HW + hw];
    const float bx = (float)wi + fx;
    const float by = (float)hi + fy;
    const float xf = floorf(bx);
    const float yf = floorf(by);
    const int   X0 = (int)xf;
    const int   Y0 = (int)yf;
    const float wx1 = bx - xf, wx0 = 1.0f - wx1;
    const float wy1 = by - yf, wy0 = 1.0f - wy1;
    const int   XW = X0 - 4, YW = Y0 - 4;   // window origin

    // ---- per-lane window slots: element e = lane + 32*s, e<100 ------------
    int   off[4];
    float msk[4];
#pragma unroll
    for (int s = 0; s < 4; ++s) {
        int e = lane + 32 * s;
        int r = e / NWIN, c = e - r * NWIN;
        int y = YW + r, x = XW + c;
        bool valid = (e < 100) && (y >= 0) && (y < HH) && (x >= 0) && (x < WW);
        int yc = min(max(y, 0), HH - 1);
        int xc = min(max(x, 0), WW - 1);
        off[s] = valid ? (yc * WW + xc) : 0;  // invalid -> alias addr 0 (L0 hit)
        msk[s] = valid ? 1.0f : 0.0f;
    }
    // dedupe invalid slot addresses onto slot0's line (cheap L0 hits)
#pragma unroll
    for (int s = 1; s < 4; ++s) if (msk[s] == 0.0f) off[s] = off[0];

    // ---- stage in1[b,:,h,w] (256 ch) into LDS -----------------------------
    float* in1s = smem + wave * 512;
    float* Dl   = smem + wave * 512 + 256;
    const float* g1 = in1 + (size_t)b * CC * HW + hw;
#pragma unroll
    for (int k = 0; k < 8; ++k)
        in1s[lane + 32 * k] = g1[(size_t)(lane + 32 * k) * HW];
    __syncthreads();

    // ---- main loop: D[slot] = sum_ch in1[ch] * in2[ch, window_slot] -------
    const float* p2 = in2 + (size_t)b * CC * HW;
    float a0 = 0.f, a1 = 0.f, a2 = 0.f, a3 = 0.f;
    const float4* in1v = (const float4*)in1s;
    for (int c4 = 0; c4 < CC / 4; ++c4) {
        float4 a = in1v[c4];                       // ds_load_b128 broadcast
        const float* pl = p2 + (size_t)c4 * 4 * HW;
        if (c4 < CC / 4 - 2) {                     // gather-ahead prefetch
            __builtin_prefetch(pl + 8 * (size_t)HW + off[0], 0, 1);
        }
#pragma unroll
        for (int j = 0; j < 4; ++j) {
            const float* pj = pl + (size_t)j * HW;
            float av = (j == 0) ? a.x : (j == 1) ? a.y : (j == 2) ? a.z : a.w;
            a0 += av * pj[off[0]];
            a1 += av * pj[off[1]];
            a2 += av * pj[off[2]];
            a3 += av * pj[off[3]];
        }
    }
    a0 *= msk[0]; a1 *= msk[1]; a2 *= msk[2]; a3 *= msk[3];

    // ---- stage D into a zero-padded 16x16 tile in LDS ---------------------
    __syncthreads();
#pragma unroll
    for (int k = 0; k < 8; ++k) Dl[lane + 32 * k] = 0.0f;
    __syncthreads();
    {
        int e, r, c;
        e = lane;      r = e / NWIN; c = e - r * NWIN; Dl[r * 16 + c] = a0;
        e = lane + 32; r = e / NWIN; c = e - r * NWIN; Dl[r * 16 + c] = a1;
        e = lane + 64; r = e / NWIN; c = e - r * NWIN; Dl[r * 16 + c] = a2;
        e = lane + 96;
        if (e < 100) { r = e / NWIN; c = e - r * NWIN; Dl[r * 16 + c] = a3; }
    }
    __syncthreads();

    // ---- T = Ay * D via v_wmma_f32_16x16x4_f32 (K=16 in 4 chunks) ---------
    // Ay[m,k] = wy0 if k==m, wy1 if k==m+1, else 0  (rows m>=9 zeroed)
    // A layout (16x4 f32): lanes0-15 rows M; VGPR0 holds K=k0 / k0+2,
    //                      VGPR1 holds K=k0+1 / k0+3 across lane halves.
    // B layout (4x16 f32): lanes = N columns, mirrored K striping.
    const int  nn = lane & 15;
    const bool hiHalf = lane >= 16;
    v8f t = {0.f, 0.f, 0.f, 0.f, 0.f, 0.f, 0.f, 0.f};
#pragma unroll
    for (int kk = 0; kk < 4; ++kk) {
        int kb = 4 * kk + (hiHalf ? 2 : 0);
        float alo = (nn < 9) ? ((kb     == nn) ? wy0 : (kb     == nn + 1) ? wy1 : 0.f) : 0.f;
        float ahi = (nn < 9) ? ((kb + 1 == nn) ? wy0 : (kb + 1 == nn + 1) ? wy1 : 0.f) : 0.f;
        v2f A = {alo, ahi};
        v2f Bv = {Dl[kb * 16 + nn], Dl[(kb + 1) * 16 + nn]};
        t = __builtin_amdgcn_wmma_f32_16x16x4_f32(
                /*neg_a=*/false, A, /*neg_b=*/false, Bv,
                /*c_mod=*/(short)0, t, /*reuse_a=*/false, /*reuse_b=*/false);
    }

    // ---- spill T (C/D layout: M = v + 8*hiHalf, N = nn) into LDS ----------
    __syncthreads();
#pragma unroll
    for (int v = 0; v < 8; ++v)
        Dl[(v + (hiHalf ? 8 : 0)) * 16 + nn] = t[v];
    __syncthreads();

    // ---- x-combine + store: corr[oy,ox] = wx0*T[oy,ox] + wx1*T[oy,ox+1] ---
#pragma unroll
    for (int s = 0; s < 3; ++s) {
        int o = lane + 32 * s;
        if (o < NPATCH * NPATCH) {
            int oy = o / NPATCH, ox = o - oy * NPATCH;
            float val = wx0 * Dl[oy * 16 + ox] + wx1 * Dl[oy * 16 + ox + 1];
            out[((size_t)(b * NPATCH * NPATCH + o)) * HW + hw] = val;
        }
    }
}

extern "C" void kernel_launch(void* const* d_in, const int* in_sizes, int n_in,
                              void* d_out, int out_size, void* d_ws, size_t ws_size,
                              hipStream_t stream) {
    const float* in1  = (const float*)d_in[0];
    const float* in2  = (const float*)d_in[1];
    const float* flow = (const float*)d_in[2];
    float* out = (float*)d_out;
    (void)in_sizes; (void)n_in; (void)out_size; (void)d_ws; (void)ws_size;

    const int npix   = BB * HH * WW;          // 65536 pixels
    const int blocks = npix / 8;              // 8 waves (pixels) per block
    corr_flow_kernel<<<blocks, 256, 0, stream>>>(in1, in2, flow, out);
}